// MLP_soft_IHT_23270132810500
// MI455X (gfx1250) — compile-verified
//
#include <hip/hip_runtime.h>
#include <hip/hip_bf16.h>

typedef float v2f __attribute__((ext_vector_type(2)));
typedef float v8f __attribute__((ext_vector_type(8)));

#define M_MEAS  256
#define N_DICT  512
#define BATCH   2048
#define S_SPARSE 50
#define ETA     0.5f
#define INV_TAU 10.0f   // 1 / 0.1

// ds_swizzle XOR patterns (group-of-32 mode: offset[14:10]=xor, [4:0]=and)
#define SWZ(x, imm) __int_as_float(__builtin_amdgcn_ds_swizzle(__float_as_int(x), (imm)))

__device__ __forceinline__ float wave32_max(float x) {
    x = fmaxf(x, SWZ(x, 0x041f));   // xor 1
    x = fmaxf(x, SWZ(x, 0x081f));   // xor 2
    x = fmaxf(x, SWZ(x, 0x101f));   // xor 4
    x = fmaxf(x, SWZ(x, 0x201f));   // xor 8
    x = fmaxf(x, SWZ(x, 0x401f));   // xor 16
    return x;
}

__device__ __forceinline__ float wave32_sum(float x) {
    x += SWZ(x, 0x041f);
    x += SWZ(x, 0x081f);
    x += SWZ(x, 0x101f);
    x += SWZ(x, 0x201f);
    x += SWZ(x, 0x401f);
    return x;
}

// ---------------------------------------------------------------------------
// Kernel 1: G = I - eta * A^T A   (N x N row-major). A is (M x N) row-major.
// One wave per 16x16 tile of G; 1024 tiles; K=256 in steps of 4 (f32 WMMA).
// ---------------------------------------------------------------------------
__global__ __launch_bounds__(256) void k_gemm_G(const float* __restrict__ A,
                                                float* __restrict__ G)
{
    const int wave = (blockIdx.x << 3) | (threadIdx.x >> 5);  // 0..1023
    const int ti   = wave >> 5;                               // row tile (N)
    const int tj   = wave & 31;                               // col tile (N)
    const int lane = threadIdx.x & 31;
    const int lo   = lane & 15;
    const int hi   = lane >> 4;

    const int colA = ti * 16 + lo;   // A-frag: lane row M = lo (over N-dim)
    const int colB = tj * 16 + lo;   // B-frag: lane col N = lo

    v8f acc = {0.f, 0.f, 0.f, 0.f, 0.f, 0.f, 0.f, 0.f};
#pragma unroll 4
    for (int k0 = 0; k0 < M_MEAS; k0 += 4) {
        const int kA = k0 + (hi << 1);     // K = 2*hi + v
        v2f a, b;
        a.x = A[(kA + 0) * N_DICT + colA];
        a.y = A[(kA + 1) * N_DICT + colA];
        b.x = A[(kA + 0) * N_DICT + colB];
        b.y = A[(kA + 1) * N_DICT + colB];
        acc = __builtin_amdgcn_wmma_f32_16x16x4_f32(false, a, false, b,
                                                    (short)0, acc, false, false);
    }
#pragma unroll
    for (int v = 0; v < 8; ++v) {
        const int i = ti * 16 + (hi << 3) + v;   // D row M = v + 8*hi
        const int j = tj * 16 + lo;              // D col N = lo
        G[i * N_DICT + j] = ((i == j) ? 1.0f : 0.0f) - ETA * acc[v];
    }
}

// ---------------------------------------------------------------------------
// Kernel 2: CT (B x N, row-major) where CT[b][n] = eta * sum_m A[m][n]*Y[b][m]
// D-tile = (A^T tile) x (Y^T tile); 4096 tiles; K=256.
// ---------------------------------------------------------------------------
__global__ __launch_bounds__(256) void k_gemm_C(const float* __restrict__ A,
                                                const float* __restrict__ Y,
                                                float* __restrict__ CT)
{
    const int wave = (blockIdx.x << 3) | (threadIdx.x >> 5);  // 0..4095
    const int ti   = wave >> 7;                               // N tile (0..31)
    const int tb   = wave & 127;                              // batch tile (0..127)
    const int lane = threadIdx.x & 31;
    const int lo   = lane & 15;
    const int hi   = lane >> 4;

    const int colA = ti * 16 + lo;
    const int rowY = tb * 16 + lo;

    v8f acc = {0.f, 0.f, 0.f, 0.f, 0.f, 0.f, 0.f, 0.f};
#pragma unroll 4
    for (int k0 = 0; k0 < M_MEAS; k0 += 4) {
        const int kA = k0 + (hi << 1);
        v2f a, b;
        a.x = A[(kA + 0) * N_DICT + colA];
        a.y = A[(kA + 1) * N_DICT + colA];
        const float2 yv = *reinterpret_cast<const float2*>(&Y[rowY * M_MEAS + kA]);
        b.x = yv.x;
        b.y = yv.y;
        acc = __builtin_amdgcn_wmma_f32_16x16x4_f32(false, a, false, b,
                                                    (short)0, acc, false, false);
    }
    // D(M = n-dim, N = batch) -> CT[batch][n]; 8 contiguous floats per lane
    float* dst = &CT[(tb * 16 + lo) * N_DICT + ti * 16 + (hi << 3)];
    float4 s0 = make_float4(ETA * acc[0], ETA * acc[1], ETA * acc[2], ETA * acc[3]);
    float4 s1 = make_float4(ETA * acc[4], ETA * acc[5], ETA * acc[6], ETA * acc[7]);
    *reinterpret_cast<float4*>(dst)     = s0;
    *reinterpret_cast<float4*>(dst + 4) = s1;
}

// ---------------------------------------------------------------------------
// Kernel 3: HT = (G @ X + C)^T, all stored (B x N) row-major (XT, CT, HT).
// 4096 tiles; K = 512; accumulator seeded from CT tile.
// ---------------------------------------------------------------------------
__global__ __launch_bounds__(256) void k_gemm_H(const float* __restrict__ G,
                                                const float* __restrict__ XT,
                                                const float* __restrict__ CT,
                                                float* __restrict__ HT)
{
    const int wave = (blockIdx.x << 3) | (threadIdx.x >> 5);
    const int ti   = wave >> 7;
    const int tb   = wave & 127;
    const int lane = threadIdx.x & 31;
    const int lo   = lane & 15;
    const int hi   = lane >> 4;

    const float* csrc = &CT[(tb * 16 + lo) * N_DICT + ti * 16 + (hi << 3)];
    const float4 c0 = *reinterpret_cast<const float4*>(csrc);
    const float4 c1 = *reinterpret_cast<const float4*>(csrc + 4);
    v8f acc;
    acc[0] = c0.x; acc[1] = c0.y; acc[2] = c0.z; acc[3] = c0.w;
    acc[4] = c1.x; acc[5] = c1.y; acc[6] = c1.z; acc[7] = c1.w;

    const float* grow = &G[(ti * 16 + lo) * N_DICT];   // A-frag rows of G
    const float* xrow = &XT[(tb * 16 + lo) * N_DICT];  // B-frag: X[k][b] = XT[b][k]
#pragma unroll 4
    for (int k0 = 0; k0 < N_DICT; k0 += 4) {
        const int kA = k0 + (hi << 1);
        const float2 gv = *reinterpret_cast<const float2*>(&grow[kA]);
        const float2 xv = *reinterpret_cast<const float2*>(&xrow[kA]);
        v2f a, b;
        a.x = gv.x; a.y = gv.y;
        b.x = xv.x; b.y = xv.y;
        acc = __builtin_amdgcn_wmma_f32_16x16x4_f32(false, a, false, b,
                                                    (short)0, acc, false, false);
    }
    float* dst = &HT[(tb * 16 + lo) * N_DICT + ti * 16 + (hi << 3)];
    float4 s0 = make_float4(acc[0], acc[1], acc[2], acc[3]);
    float4 s1 = make_float4(acc[4], acc[5], acc[6], acc[7]);
    *reinterpret_cast<float4*>(dst)     = s0;
    *reinterpret_cast<float4*>(dst + 4) = s1;
}

// ---------------------------------------------------------------------------
// Kernel 4: soft top-s mask + threshold. One 512-thread block per sample b.
// Sort + scan use LDS (once per sample). The hot 50-row softmax loop uses
// wave32 ds_swizzle reductions: 2 barriers/row instead of ~18.
// ---------------------------------------------------------------------------
__global__ __launch_bounds__(512) void k_mask(const float* __restrict__ HT,
                                              const float* __restrict__ W,
                                              float* __restrict__ Xout)
{
    __shared__ float sval[N_DICT];
    __shared__ int   sidx[N_DICT];
    __shared__ float sB[N_DICT];
    __shared__ float sred[N_DICT];
    __shared__ float pmax[16];
    __shared__ float psum[16];

    const int n    = threadIdx.x;
    const int b    = blockIdx.x;
    const int lane = n & 31;
    const int wid  = n >> 5;

    const float h = HT[b * N_DICT + n];
    const float z = fabsf(W[n] * h);

    sval[n] = z;
    sidx[n] = n;
    __syncthreads();

    // Bitonic sort ascending on (sval, sidx). Ties harmless: B depends on value only.
    for (int k = 2; k <= N_DICT; k <<= 1) {
        for (int j = k >> 1; j > 0; j >>= 1) {
            const int ixj = n ^ j;
            if (ixj > n) {
                const bool up = ((n & k) == 0);
                const float a = sval[n];
                const float c = sval[ixj];
                if ((a > c) == up) {
                    sval[n] = c; sval[ixj] = a;
                    const int t = sidx[n]; sidx[n] = sidx[ixj]; sidx[ixj] = t;
                }
            }
            __syncthreads();
        }
    }

    // Inclusive scan of sorted values (Hillis-Steele).
    sred[n] = sval[n];
    __syncthreads();
    for (int off = 1; off < N_DICT; off <<= 1) {
        const float t = (n >= off) ? sred[n - off] : 0.0f;
        __syncthreads();
        sred[n] += t;
        __syncthreads();
    }
    const float total = sred[N_DICT - 1];
    // Bs_k = (2(k+1)-N)*zs_k - 2*cs_k + total, scattered to original index.
    const float Bs = (2.0f * (float)(n + 1) - (float)N_DICT) * sval[n]
                     - 2.0f * sred[n] + total;
    sB[sidx[n]] = Bs;
    __syncthreads();
    const float Bv = sB[n];
    __syncthreads();

    // mask = sum_{i=1..s} softmax_n( ((N+1-2i)*z - B)/tau )
    const float u  = z  * INV_TAU;
    const float vb = Bv * INV_TAU;
    float mask = 0.0f;
    for (int i = 1; i <= S_SPARSE; ++i) {
        const float coef  = (float)(N_DICT + 1 - 2 * i);
        const float logit = fmaf(coef, u, -vb);

        // per-row max: wave-local swizzle reduce, then combine 16 partials
        const float wm = wave32_max(logit);
        if (lane == 0) pmax[wid] = wm;
        __syncthreads();
        float m = pmax[0];
#pragma unroll
        for (int w = 1; w < 16; ++w) m = fmaxf(m, pmax[w]);

        const float e = __expf(logit - m);

        const float ws = wave32_sum(e);
        if (lane == 0) psum[wid] = ws;
        __syncthreads();
        float s = psum[0];
#pragma unroll
        for (int w = 1; w < 16; ++w) s += psum[w];

        mask += e / s;
    }

    Xout[b * N_DICT + n] = mask * h;
}

// ---------------------------------------------------------------------------
// Host-side orchestration (graph-capture safe: only kernel launches on stream)
// ---------------------------------------------------------------------------
extern "C" void kernel_launch(void* const* d_in, const int* in_sizes, int n_in,
                              void* d_out, int out_size, void* d_ws, size_t ws_size,
                              hipStream_t stream) {
    const float* Y = (const float*)d_in[0];   // (2048, 256)
    const float* A = (const float*)d_in[1];   // (256, 512)
    const float* W = (const float*)d_in[2];   // (512,)
    float* out = (float*)d_out;               // (2048, 512) == X^T

    char* ws = (char*)d_ws;
    float* G  = (float*)(ws);                           // 512*512*4   = 1 MB
    float* CT = (float*)(ws + (size_t)(1 << 20));       // 2048*512*4  = 4 MB
    float* XT = (float*)(ws + (size_t)(5 << 20));       // 4 MB
    float* HT = (float*)(ws + (size_t)(9 << 20));       // 4 MB

    // Precompute G and C^T.
    k_gemm_G<<<128, 256, 0, stream>>>(A, G);            // 1024 tiles / 8 waves
    k_gemm_C<<<512, 256, 0, stream>>>(A, Y, CT);        // 4096 tiles / 8 waves

    // Layer 1: X = 0  =>  H = C.
    k_mask<<<BATCH, N_DICT, 0, stream>>>(CT, W, XT);

    // Layers 2..3.
    for (int l = 0; l < 2; ++l) {
        k_gemm_H<<<512, 256, 0, stream>>>(G, XT, CT, HT);
        k_mask<<<BATCH, N_DICT, 0, stream>>>(HT, W, XT);
    }

    // Layer 4: final mask writes X^T straight into d_out.
    k_gemm_H<<<512, 256, 0, stream>>>(G, XT, CT, HT);
    k_mask<<<BATCH, N_DICT, 0, stream>>>(HT, W, out);
}